// HardPositiveInfoNCELoss_86414741995690
// MI455X (gfx1250) — compile-verified
//
#include <hip/hip_runtime.h>
#include <hip/hip_bf16.h>
#include <math.h>

// Problem constants (from reference setup_inputs)
#define BQ   64      // B
#define QN   32      // Q
#define CN   64      // C
#define SN   256     // S
#define DN   128     // D
#define TAU  0.02f

typedef __attribute__((ext_vector_type(2))) float v2f;
typedef __attribute__((ext_vector_type(8))) float v8f;

// One workgroup per (b,c). 8 waves; each wave: 1 m-tile x 4 n-tiles of 16x16,
// K looped in steps of 4 via V_WMMA_F32_16X16X4_F32. Row-max epilogue is done
// in-register on the WMMA accumulators + 16-lane shuffle butterflies.
__global__ __launch_bounds__(256) void
sim_tile_kernel(const float* __restrict__ qemb,   // [B,Q,D]
                const float* __restrict__ demb,   // [C,S,D]
                const int*   __restrict__ mids,   // [C,S]
                const int*   __restrict__ qtypes, // [B]
                const float* __restrict__ qmask,  // [B,Q]
                float*       __restrict__ sim)    // [B,C]
{
    __shared__ int   smod[SN];        // modality ids of doc c
    __shared__ float pmax[8 * 16];    // per-wave per-row partial max
    __shared__ float pmsk[8 * 16];    // per-wave per-row masked partial max
    __shared__ float red[2 * QN];     // final per-row maxes

    const int b    = blockIdx.y;
    const int c    = blockIdx.x;
    const int tid  = threadIdx.x;
    const int w    = tid >> 5;
    const int lane = tid & 31;
    const int half = lane >> 4;       // 0: lanes 0-15, 1: lanes 16-31
    const int l    = lane & 15;

    const int mtile = w & 1;          // rows m0..m0+15
    const int ngrp  = w >> 1;         // 4 n-tiles each
    const int m0    = mtile << 4;
    const int n0    = (ngrp << 2) << 4;

    // Stage modality ids for doc c.
    if (tid < SN) smod[tid] = mids[c * SN + tid];
    __syncthreads();

    const float* qb = qemb + (size_t)b * QN * DN;
    const float* db = demb + (size_t)c * SN * DN;

    // Per-lane base pointers (k offset 2*half per fp32 WMMA A/B layout).
    const float* qrow = qb + (m0 + l) * DN + (half << 1);
    const float* dr0  = db + (n0 +  0 + l) * DN + (half << 1);
    const float* dr1  = db + (n0 + 16 + l) * DN + (half << 1);
    const float* dr2  = db + (n0 + 32 + l) * DN + (half << 1);
    const float* dr3  = db + (n0 + 48 + l) * DN + (half << 1);

    v8f acc0 = {}, acc1 = {}, acc2 = {}, acc3 = {};

    #pragma unroll 4
    for (int k0 = 0; k0 < DN; k0 += 4) {
        v2f a  = *(const v2f*)(qrow + k0);
        v2f b0 = *(const v2f*)(dr0  + k0);
        v2f b1 = *(const v2f*)(dr1  + k0);
        v2f b2 = *(const v2f*)(dr2  + k0);
        v2f b3 = *(const v2f*)(dr3  + k0);
        // (neg_a, A, neg_b, B, c_mod, C, reuse_a, reuse_b)
        acc0 = __builtin_amdgcn_wmma_f32_16x16x4_f32(false, a, false, b0,
                                                     (short)0, acc0, false, false);
        acc1 = __builtin_amdgcn_wmma_f32_16x16x4_f32(false, a, false, b1,
                                                     (short)0, acc1, false, false);
        acc2 = __builtin_amdgcn_wmma_f32_16x16x4_f32(false, a, false, b2,
                                                     (short)0, acc2, false, false);
        acc3 = __builtin_amdgcn_wmma_f32_16x16x4_f32(false, a, false, b3,
                                                     (short)0, acc3, false, false);
    }

    // In-register epilogue. Accumulator VGPR r holds row (m0+r) in lanes 0-15
    // and row (m0+r+8) in lanes 16-31; each lane's column is s = n0 + 16j + l.
    const int qt  = qtypes[b];
    const int md0 = smod[n0 +  0 + l];
    const int md1 = smod[n0 + 16 + l];
    const int md2 = smod[n0 + 32 + l];
    const int md3 = smod[n0 + 48 + l];

    #pragma unroll
    for (int r = 0; r < 8; ++r) {
        float v0 = acc0[r], v1 = acc1[r], v2 = acc2[r], v3 = acc3[r];
        float mx = fmaxf(fmaxf(v0, v1), fmaxf(v2, v3));
        float mm = (md0 == qt) ? v0 : -INFINITY;
        if (md1 == qt) mm = fmaxf(mm, v1);
        if (md2 == qt) mm = fmaxf(mm, v2);
        if (md3 == qt) mm = fmaxf(mm, v3);
        // 16-lane butterfly: halves (rows) reduce independently.
        #pragma unroll
        for (int off = 1; off < 16; off <<= 1) {
            mx = fmaxf(mx, __shfl_xor(mx, off, 32));
            mm = fmaxf(mm, __shfl_xor(mm, off, 32));
        }
        if (l == 0) {
            const int rowLocal = r + (half << 3);   // 0..15 within the m-tile
            pmax[w * 16 + rowLocal] = mx;
            pmsk[w * 16 + rowLocal] = mm;
        }
    }
    __syncthreads();

    // Combine the 4 waves that share each m-tile (waves mt, mt+2, mt+4, mt+6).
    if (tid < QN) {
        const int q  = tid;
        const int mt = q >> 4;
        const int rr = q & 15;
        float mx = -INFINITY, mm = -INFINITY;
        #pragma unroll
        for (int g = 0; g < 4; ++g) {
            const int w2 = mt + (g << 1);
            mx = fmaxf(mx, pmax[w2 * 16 + rr]);
            mm = fmaxf(mm, pmsk[w2 * 16 + rr]);
        }
        red[q]      = mx;
        red[QN + q] = mm;
    }
    __syncthreads();

    if (tid == 0) {
        float qlen = 0.0f;
        for (int i = 0; i < QN; ++i) qlen += qmask[b * QN + i];
        qlen = fmaxf(qlen, 1.0f);
        float s_all = 0.0f, s_msk = 0.0f;
        for (int i = 0; i < QN; ++i) { s_all += red[i]; s_msk += red[QN + i]; }
        // Diagonal entry gets the per-modality masked score (sim_fixed).
        sim[b * CN + c] = ((b == c) ? s_msk : s_all) / qlen;
    }
}

// Single block, 64 threads: per-row log-softmax over C, mean of -logp[b,b].
__global__ __launch_bounds__(64) void
infonce_loss_kernel(const float* __restrict__ sim, float* __restrict__ out)
{
    __shared__ float red[BQ];
    const int t = threadIdx.x;
    const float inv_tau = 1.0f / TAU;

    const float* row = sim + t * CN;
    float m = -INFINITY;
    for (int c = 0; c < CN; ++c) m = fmaxf(m, row[c] * inv_tau);
    float se = 0.0f;
    for (int c = 0; c < CN; ++c) se += expf(row[c] * inv_tau - m);
    const float logp = row[t] * inv_tau - m - logf(se);
    red[t] = -logp;
    __syncthreads();

    if (t == 0) {
        float s = 0.0f;
        for (int i = 0; i < BQ; ++i) s += red[i];
        out[0] = s / (float)BQ;
    }
}

extern "C" void kernel_launch(void* const* d_in, const int* in_sizes, int n_in,
                              void* d_out, int out_size, void* d_ws, size_t ws_size,
                              hipStream_t stream)
{
    const float* qemb   = (const float*)d_in[0];  // [64,32,128] f32
    const float* demb   = (const float*)d_in[1];  // [64,256,128] f32
    const int*   mids   = (const int*)  d_in[2];  // [64,256] i32
    const int*   qtypes = (const int*)  d_in[3];  // [64] i32
    const float* qmask  = (const float*)d_in[4];  // [64,32] f32

    float* sim = (float*)d_ws;                    // [64,64] f32 scratch

    dim3 grid(CN, BQ);
    sim_tile_kernel<<<grid, 256, 0, stream>>>(qemb, demb, mids, qtypes, qmask, sim);
    infonce_loss_kernel<<<1, 64, 0, stream>>>(sim, (float*)d_out);
}